// Kernel_16550548659508808100_53472342835837
// MI455X (gfx1250) — compile-verified
//
#include <hip/hip_runtime.h>
#include <math.h>

// ---------------------------------------------------------------------------
// Shapes: n=16, c=112, h=48, w=48, hw=2304.
// Algebra: t10 = -x (t8 cancels), rolls in t1..t6 cancel, and
//   out[n,c,q] = (1/48) * ( 2*A_s[n,c] - (1/sqrt(112)) * sum_ch M1[n,c,ch]*t13[n,ch,q] )
//   M1[n]  = t19[n] (112 x 2304) @ x[n]^T (2304 x 112)      -> WMMA f32 16x16x4
//   out[n] = M1[n]  (112 x 112)  @ t13[n] (112 x 2304)      -> WMMA f32 16x16x4
//   A_s[n,c] = sum_p t19[n,c,p] * t7[n, p/48]
// ---------------------------------------------------------------------------

#define NB 16
#define CC 112
#define HH 48
#define WWD 48
#define HWSZ 2304          // HH*WWD
#define CHW (CC*HWSZ)      // 258048

typedef __attribute__((ext_vector_type(2))) float v2f;
typedef __attribute__((ext_vector_type(8))) float v8f;

__device__ __forceinline__ v8f wmma4(v2f a, v2f b, v8f c) {
  // D = A(16x4,f32) * B(4x16,f32) + C(16x16,f32)
  return __builtin_amdgcn_wmma_f32_16x16x4_f32(false, a, false, b, (short)0, c,
                                               false, false);
}

// ---- xbar[n,h,w] = mean over c of x -------------------------------------
__global__ void xbar_kernel(const float* __restrict__ x, float* __restrict__ xbar) {
  int idx = blockIdx.x * blockDim.x + threadIdx.x;
  if (idx >= NB * HWSZ) return;
  int n = idx / HWSZ, r = idx % HWSZ;
  const float* p = x + (size_t)n * CHW + r;
  float s = 0.f;
#pragma unroll 4
  for (int c = 0; c < CC; ++c) s += p[(size_t)c * HWSZ];
  xbar[idx] = s * (1.0f / CC);
}

// ---- t7[n,h] = sum_{k,w} p7[k,w] * xbar[n,h, w+3k-3] (zero pad) ----------
__global__ void t7_kernel(const float* __restrict__ xbar,
                          const float* __restrict__ p7,
                          float* __restrict__ t7) {
  int idx = blockIdx.x * blockDim.x + threadIdx.x;
  if (idx >= NB * HH) return;
  int n = idx / HH, h = idx % HH;
  const float* row = xbar + n * HWSZ + h * WWD;
  float s = 0.f;
  for (int k = 0; k < 3; ++k)
    for (int w = 0; w < WWD; ++w) {
      int col = w + 3 * k - 3;
      if (col >= 0 && col < WWD) s += p7[k * WWD + w] * row[col];
    }
  t7[idx] = s;
}

// ---- t17[n,k] = (1/48) sum_i |t7[n, 2k+i-6]| (zero pad) ------------------
__global__ void t17_kernel(const float* __restrict__ t7, float* __restrict__ t17) {
  int idx = blockIdx.x * blockDim.x + threadIdx.x;
  if (idx >= NB * 7) return;
  int n = idx / 7, k = idx % 7;
  float s = 0.f;
  for (int i = 0; i < HH; ++i) {
    int j = 2 * k + i - 6;
    if (j >= 0 && j < HH) s += fabsf(t7[n * HH + j]);
  }
  t17[idx] = s * (1.0f / HH);
}

// ---- t19 = p19[c] * (t17[n, flat%7] + x) ---------------------------------
__global__ void t19_kernel(const float* __restrict__ x,
                           const float* __restrict__ p19,
                           const float* __restrict__ t17,
                           float* __restrict__ t19) {
  int idx = blockIdx.x * blockDim.x + threadIdx.x;
  if (idx >= NB * CHW) return;
  int n = idx / CHW, rem = idx % CHW;
  int c = rem / HWSZ;
  t19[idx] = p19[c] * (t17[n * 7 + rem % 7] + x[idx]);
}

// ---- t13 = max( conv(-x) , x ): 1x3 kernel, w-dilation 2, pad 2 ----------
// one block per (n,h); padded row of all 112 channels cached in LDS.
__global__ void conv_t13_kernel(const float* __restrict__ x,
                                const float* __restrict__ cw,
                                float* __restrict__ t13) {
  __shared__ float xr[CC * 52];              // 52 = 48 + 2 pad each side
  int b = blockIdx.x;
  int n = b / HH, h = b % HH;
  const float* xrow = x + (size_t)n * CHW + h * WWD;
  for (int i = threadIdx.x; i < CC * 52; i += blockDim.x) {
    int ch = i / 52, col = i % 52;
    xr[i] = (col >= 2 && col < 50) ? xrow[(size_t)ch * HWSZ + (col - 2)] : 0.f;
  }
  __syncthreads();
  int w = threadIdx.x % WWD;                 // blockDim = 192 = 4*48
  int o0 = threadIdx.x / WWD;                // 0..3
  for (int o = o0; o < CC; o += 4) {
    float acc = 0.f;
    const float* wp = cw + (size_t)o * CC * 3;
    for (int i = 0; i < CC; ++i) {
      const float* xp = xr + i * 52 + w;     // taps at +0,+2,+4 (== w+2k-2 padded)
      acc = fmaf(wp[i * 3 + 0], xp[0], acc);
      acc = fmaf(wp[i * 3 + 1], xp[2], acc);
      acc = fmaf(wp[i * 3 + 2], xp[4], acc);
    }
    float xv = xrow[(size_t)o * HWSZ + w];
    t13[(size_t)n * CHW + (size_t)o * HWSZ + h * WWD + w] = fmaxf(-acc, xv);
  }
}

// ---- M1[n] = t19[n] (112x2304) @ x[n]^T : one wave per 16x16 tile --------
__global__ void m1_kernel(const float* __restrict__ t19,
                          const float* __restrict__ x,
                          float* __restrict__ m1) {
  int wave = (blockIdx.x * blockDim.x + threadIdx.x) >> 5;
  int lane = threadIdx.x & 31;
  if (wave >= NB * 49) return;
  int n = wave / 49, t = wave % 49, tm = t / 7, tn = t % 7;
  int r = lane & 15;
  int kh = (lane >> 4) << 1;                 // K sub-offset: 0 or 2
  const float* A = t19 + (size_t)n * CHW + (size_t)(tm * 16 + r) * HWSZ + kh;
  const float* B = x   + (size_t)n * CHW + (size_t)(tn * 16 + r) * HWSZ + kh;
  v8f acc = {0.f, 0.f, 0.f, 0.f, 0.f, 0.f, 0.f, 0.f};
#pragma unroll 8
  for (int k = 0; k < HWSZ; k += 4) {
    v2f a = *(const v2f*)(A + k);
    v2f b = *(const v2f*)(B + k);
    acc = wmma4(a, b, acc);
  }
  int mbase = tm * 16 + ((lane >> 4) << 3);  // M = v + 8*(lane>=16)
  int col = tn * 16 + r;
#pragma unroll
  for (int v = 0; v < 8; ++v)
    m1[(size_t)n * CC * CC + (size_t)(mbase + v) * CC + col] = acc[v];
}

// ---- A_s[n,c] = sum_{h,w} t19[n,c,h,w] * t7[n,h] -------------------------
__global__ void asum_kernel(const float* __restrict__ t19,
                            const float* __restrict__ t7,
                            float* __restrict__ asum) {
  __shared__ float red[256];
  int b = blockIdx.x;
  int n = b / CC, c = b % CC;
  const float* base = t19 + (size_t)n * CHW + (size_t)c * HWSZ;
  const float* t7n = t7 + n * HH;
  float s = 0.f;
  for (int i = threadIdx.x; i < HWSZ; i += blockDim.x)
    s += base[i] * t7n[i / WWD];
  red[threadIdx.x] = s;
  __syncthreads();
  for (int off = 128; off > 0; off >>= 1) {
    if ((int)threadIdx.x < off) red[threadIdx.x] += red[threadIdx.x + off];
    __syncthreads();
  }
  if (threadIdx.x == 0) asum[b] = red[0];
}

// ---- out[n] = (1/48)*(2*A_s - (1/sqrt(112)) * M1[n] @ t13[n]) ------------
__global__ void out_kernel(const float* __restrict__ m1,
                           const float* __restrict__ t13,
                           const float* __restrict__ asum,
                           float* __restrict__ out) {
  int wave = (blockIdx.x * blockDim.x + threadIdx.x) >> 5;
  int lane = threadIdx.x & 31;
  if (wave >= NB * 7 * 144) return;
  int n = wave / 1008, t = wave % 1008, tm = t / 144, tq = t % 144;
  int r = lane & 15;
  int kh = (lane >> 4) << 1;
  const float* A = m1 + (size_t)n * CC * CC + (size_t)(tm * 16 + r) * CC + kh;
  const float* B = t13 + (size_t)n * CHW + (size_t)kh * HWSZ + tq * 16 + r;
  v8f acc = {0.f, 0.f, 0.f, 0.f, 0.f, 0.f, 0.f, 0.f};
#pragma unroll
  for (int k = 0; k < CC; k += 4) {          // 28 WMMA steps
    v2f a = *(const v2f*)(A + k);
    v2f b;
    b.x = B[(size_t)k * HWSZ];
    b.y = B[(size_t)(k + 1) * HWSZ];
    acc = wmma4(a, b, acc);
  }
  const float inv_hw = 1.0f / 48.0f;                 // 1/sqrt(2304)
  const float inv_c = 0.09449111825230679f;          // 1/sqrt(112)
  int mbase = tm * 16 + ((lane >> 4) << 3);
  int q = tq * 16 + r;
#pragma unroll
  for (int v = 0; v < 8; ++v) {
    int m = mbase + v;
    out[(size_t)n * CHW + (size_t)m * HWSZ + q] =
        inv_hw * (2.0f * asum[n * CC + m] - inv_c * acc[v]);
  }
}

extern "C" void kernel_launch(void* const* d_in, const int* in_sizes, int n_in,
                              void* d_out, int out_size, void* d_ws, size_t ws_size,
                              hipStream_t stream) {
  const float* x   = (const float*)d_in[0];
  const float* p7  = (const float*)d_in[1];
  // d_in[2] = p8_w is algebraically dead: t10 = t8 - (x + t8) = -x
  const float* p19 = (const float*)d_in[3];
  const float* cw  = (const float*)d_in[4];
  float* out = (float*)d_out;

  char* ws = (char*)d_ws;
  float* xbar = (float*)(ws + 0);          //  36,864 f  (147,456 B)
  float* t7   = (float*)(ws + 147456);     //     768 f
  float* t17  = (float*)(ws + 150528);     //     112 f
  float* asum = (float*)(ws + 150976);     //   1,792 f
  float* t19  = (float*)(ws + 158144);     // 4,128,768 f (16.5 MB)
  float* t13  = (float*)(ws + 16673216);   // 4,128,768 f (16.5 MB)
  float* m1   = (float*)(ws + 33188288);   //   200,704 f -> total ~32.4 MB

  hipLaunchKernelGGL(xbar_kernel, dim3(144), dim3(256), 0, stream, x, xbar);
  hipLaunchKernelGGL(t7_kernel, dim3(3), dim3(256), 0, stream, xbar, p7, t7);
  hipLaunchKernelGGL(t17_kernel, dim3(1), dim3(128), 0, stream, t7, t17);
  hipLaunchKernelGGL(t19_kernel, dim3(16128), dim3(256), 0, stream, x, p19, t17, t19);
  hipLaunchKernelGGL(conv_t13_kernel, dim3(768), dim3(192), 0, stream, x, cw, t13);
  hipLaunchKernelGGL(m1_kernel, dim3(196), dim3(128), 0, stream, t19, x, m1);
  hipLaunchKernelGGL(asum_kernel, dim3(1792), dim3(256), 0, stream, t19, t7, asum);
  hipLaunchKernelGGL(out_kernel, dim3(4032), dim3(128), 0, stream, m1, t13, asum, out);
}